// LocalLayer_9603546874456
// MI455X (gfx1250) — compile-verified
//
#include <hip/hip_runtime.h>
#include <stdint.h>

// ---------------------------------------------------------------------------
// GCN layer: out = leaky_relu( [norm_adj^T @ xw[:62] + d^2*xw[:62] ; xw[62:]] + b )
// xw = x.reshape(-1,128) @ W,   B=8192, N=62, FIN=128, FOUT=64
//
// Memory-bound (AI ~21 flop/B, 390MB @ 23.3TB/s ~ 17us floor) -> full f32
// precision via V_WMMA_F32_16X16X4_F32; x-tiles staged to LDS with the CDNA5
// async-to-LDS path (ASYNCcnt), double-buffered across 4 tiles per block.
// k-loop is FULLY unrolled so the W-fragment array is SROA'd into registers
// (dynamic indexing was forcing it into scratch in earlier rounds).
// ---------------------------------------------------------------------------

typedef float v2f __attribute__((ext_vector_type(2)));
typedef float v8f __attribute__((ext_vector_type(8)));

#define GCN_B     8192
#define GCN_N     62
#define GCN_FIN   128
#define GCN_FOUT  64
#define GCN_ROWS  (GCN_B * GCN_N)      // 507904
#define TILE_ROWS 64                   // rows per tile
#define XPITCH    132                  // LDS row pitch in floats (conflict-free ds_load_b64)
#define ITERS     4                    // tiles per block
#define TILE_BYTES (TILE_ROWS * XPITCH * 4)   // 33792 B per LDS buffer

__global__ __launch_bounds__(128, 1)
void gcn_gemm_wmma(const float* __restrict__ x,
                   const float* __restrict__ W,
                   const float* __restrict__ b,
                   float* __restrict__ out)
{
    extern __shared__ float sX[];                 // 2 * TILE_ROWS * XPITCH floats

    const int tid   = threadIdx.x;
    const int wave  = tid >> 5;                   // 0..3 -> column tile
    const int lane  = tid & 31;
    const int lhalf = lane >> 4;                  // 0: K{0,1} half, 1: K{2,3} half
    const int l16   = lane & 15;
    const long blk0 = (long)blockIdx.x * (TILE_ROWS * ITERS);

    // ---- B fragments (W) into registers: lane holds col, rows k..k+1 (+2 hi half)
    const int col = wave * 16 + l16;
    v2f bw[32];
#pragma unroll
    for (int kk = 0; kk < 32; ++kk) {
        const int k = kk * 4 + lhalf * 2;
        bw[kk].x = W[(k    ) * GCN_FOUT + col];
        bw[kk].y = W[(k + 1) * GCN_FOUT + col];
    }
    const float bias = b[col];

    // ---- async global->LDS tile staging (CDNA5, tracked by ASYNCcnt) ----
    // Tile = 64 rows x 128 f32 = 2048 x b128. Lane handles elements
    // i = tid + 128*j (j=0..15): row = (tid>>5)+4j, c4 = tid&31.
    // LDS byte offset = row*528 + c4*16  (row pitch 132 floats = 528 B).
    // Generic shared pointer bits [31:0] == LDS byte offset (ISA 10.2).
    const uint32_t ldsBase = (uint32_t)(uintptr_t)sX;
    const uint32_t laneLds = (uint32_t)((tid >> 5) * (XPITCH * 4))   // row component: 528 B/row
                           + (uint32_t)((tid & 31) * 16);            // column component

    auto issue_tile = [&](int it, int buf) {
        const uint64_t g0 = (uint64_t)(uintptr_t)(x + (blk0 + (long)it * TILE_ROWS) * GCN_FIN)
                          + (uint64_t)tid * 16;
        const uint32_t l0 = ldsBase + (uint32_t)buf * TILE_BYTES + laneLds;
#pragma unroll
        for (int j = 0; j < 16; ++j) {
            const uint32_t laddr = l0 + (uint32_t)j * (4 * XPITCH * 4); // +4 rows: 2112 B
            const uint64_t gaddr = g0 + (uint64_t)j * (128 * 16);       // +128 lanes: 2048 B
            asm volatile("global_load_async_to_lds_b128 %0, %1, off"
                         :: "v"(laddr), "v"(gaddr)
                         : "memory");
        }
    };

    issue_tile(0, 0);

#pragma unroll 1
    for (int it = 0; it < ITERS; ++it) {
        // prefetch next tile into the other buffer, then wait for current tile
        if (it + 1 < ITERS) {
            issue_tile(it + 1, (it + 1) & 1);
            asm volatile("s_wait_asynccnt 16" ::: "memory");  // tile it's 16 loads done
        } else {
            asm volatile("s_wait_asynccnt 0" ::: "memory");
        }
        __syncthreads();                                      // LDS visible block-wide

        const float* __restrict__ sx = sX + (it & 1) * (TILE_ROWS * XPITCH);
        const long r0 = blk0 + (long)it * TILE_ROWS;

        // ---- main WMMA loop: 4 row tiles, K in steps of 4; FULL unroll so
        //      bw[kk] indices are constants (keeps bw in registers).
        v8f acc[4] = {};
#pragma unroll
        for (int kk = 0; kk < 32; ++kk) {
            const int kbase = kk * 4 + lhalf * 2;
            // A frag: lanes 0-15 row M=l16 K{k,k+1}; lanes 16-31 same row K{k+2,k+3}
            const v2f a0 = *(const v2f*)&sx[(l16     ) * XPITCH + kbase];
            const v2f a1 = *(const v2f*)&sx[(l16 + 16) * XPITCH + kbase];
            const v2f a2 = *(const v2f*)&sx[(l16 + 32) * XPITCH + kbase];
            const v2f a3 = *(const v2f*)&sx[(l16 + 48) * XPITCH + kbase];
            acc[0] = __builtin_amdgcn_wmma_f32_16x16x4_f32(false, a0, false, bw[kk],
                                                           (short)0, acc[0], false, false);
            acc[1] = __builtin_amdgcn_wmma_f32_16x16x4_f32(false, a1, false, bw[kk],
                                                           (short)0, acc[1], false, false);
            acc[2] = __builtin_amdgcn_wmma_f32_16x16x4_f32(false, a2, false, bw[kk],
                                                           (short)0, acc[2], false, false);
            acc[3] = __builtin_amdgcn_wmma_f32_16x16x4_f32(false, a3, false, bw[kk],
                                                           (short)0, acc[3], false, false);
        }

        // ---- epilogue: C/D layout: VGPR j -> M = j + 8*lhalf, N = l16
        if (r0 != 0) {
            // fast path (uniform branch): no rows < 62 in this tile
#pragma unroll
            for (int t = 0; t < 4; ++t) {
#pragma unroll
                for (int j = 0; j < 8; ++j) {
                    const long grow = r0 + t * 16 + (j + lhalf * 8);
                    float v = acc[t][j] + bias;
                    out[grow * GCN_FOUT + col] =
                        fmaxf(v, 0.0f) + 0.01f * fminf(v, 0.0f);
                }
            }
        } else {
            // first tile only: rows < 62 are written raw for kernel 2
#pragma unroll
            for (int t = 0; t < 4; ++t) {
#pragma unroll
                for (int j = 0; j < 8; ++j) {
                    const long grow = (long)(t * 16 + (j + lhalf * 8));
                    float v = acc[t][j];
                    float* dst = &out[grow * GCN_FOUT + col];
                    if (grow < GCN_N) {
                        *dst = v;
                    } else {
                        v += bias;
                        *dst = fmaxf(v, 0.0f) + 0.01f * fminf(v, 0.0f);
                    }
                }
            }
        }

        __syncthreads();   // all waves done reading buf(it&1) before it is re-filled
    }
}

// ---------------------------------------------------------------------------
// Kernel 2: fix up rows 0..61 in place:
//   out[i,j] = leaky( sum_k adj[k,i]*d[k]*d[i]*xw[k,j] + d[i]^2*xw[i,j] + b[j] )
// Single block; reads staged to LDS before any write.
// ---------------------------------------------------------------------------
__global__ __launch_bounds__(256)
void gcn_first_rows(const float* __restrict__ adj,
                    const float* __restrict__ b,
                    float* __restrict__ out)
{
    __shared__ float sXW[GCN_N * GCN_FOUT];       // 62*64
    __shared__ float sAdj[GCN_N * GCN_N];         // 62*62
    __shared__ float sD[GCN_N];

    const int tid = threadIdx.x;
    for (int i = tid; i < GCN_N * GCN_FOUT; i += 256) sXW[i]  = out[i];
    for (int i = tid; i < GCN_N * GCN_N;    i += 256) sAdj[i] = adj[i];
    __syncthreads();

    if (tid < GCN_N) {
        float s = 1.0f;                            // deg = 1 + adj.sum(axis=0)
        for (int k = 0; k < GCN_N; ++k) s += sAdj[k * GCN_N + tid];
        sD[tid] = rsqrtf(s);
    }
    __syncthreads();

    for (int idx = tid; idx < GCN_N * GCN_FOUT; idx += 256) {
        const int i = idx >> 6;                    // node row
        const int j = idx & 63;                    // feature col
        const float di = sD[i];
        float s = di * di * sXW[i * GCN_FOUT + j];
        for (int k = 0; k < GCN_N; ++k)
            s += sAdj[k * GCN_N + i] * sD[k] * di * sXW[k * GCN_FOUT + j];
        s += b[j];
        out[idx] = fmaxf(s, 0.0f) + 0.01f * fminf(s, 0.0f);
    }
}

// ---------------------------------------------------------------------------
extern "C" void kernel_launch(void* const* d_in, const int* in_sizes, int n_in,
                              void* d_out, int out_size, void* d_ws, size_t ws_size,
                              hipStream_t stream)
{
    const float* x   = (const float*)d_in[0];
    const float* adj = (const float*)d_in[1];
    const float* W   = (const float*)d_in[2];
    const float* b   = (const float*)d_in[3];
    float* out = (float*)d_out;

    const int blocks = GCN_ROWS / (TILE_ROWS * ITERS);       // 1984, divides evenly
    const size_t smem = 2 * TILE_BYTES;                      // 67584 B dynamic LDS

    gcn_gemm_wmma<<<blocks, 128, smem, stream>>>(x, W, b, out);
    gcn_first_rows<<<1, 256, 0, stream>>>(adj, b, out);
}